// Dilution_15040975470785
// MI455X (gfx1250) — compile-verified
//
#include <hip/hip_runtime.h>

// Problem geometry (from the reference): 2x nearest "dilution" scatter.
#define HF 512
#define WF 512
#define HT 1024
#define WT 1024
#define BC 24              // B*C = 8*3
#define ROWS_PER_BLOCK 8   // input rows per block
#define STAGES 4           // each stage = 2 input rows = 1024 floats = 4 KiB
#define THREADS 256        // 8 wave32s

// ---- CDNA5 async global->LDS path (gfx1250) -------------------------------
// VDST of global_load_async_to_lds_* holds the per-lane LDS *byte* address
// (relative to the workgroup LDS base). An addrspace(3) pointer's integer
// value is exactly that byte offset.
typedef __attribute__((address_space(3))) const void lds_cvoid;

__device__ __forceinline__ unsigned lds_byte_addr(const void* p) {
    return (unsigned)(unsigned long long)(lds_cvoid*)p;
}

__device__ __forceinline__ void async_copy_b128(unsigned lds_off, const float* gaddr) {
    // 16 bytes per lane: global memory -> LDS, tracked by ASYNCcnt.
    asm volatile("global_load_async_to_lds_b128 %0, %1, off"
                 :: "v"(lds_off), "v"(gaddr)
                 : "memory");
}

// Wait until ASYNCcnt <= n (async loads complete in order).
#define WAIT_ASYNCCNT(n) asm volatile("s_wait_asynccnt " #n ::: "memory")
// ---------------------------------------------------------------------------

__global__ __launch_bounds__(THREADS)
void dilution_2x_kernel(const float* __restrict__ x, float* __restrict__ out) {
    // Double-buffered staging: 2 buffers x 256 lanes x 16B = 8 KiB LDS.
    __shared__ float4 smem[2][THREADS];

    const int tid   = threadIdx.x;
    const int bc    = blockIdx.x >> 6;        // / (HF / ROWS_PER_BLOCK)
    const int strip = blockIdx.x & 63;
    const int row0  = strip * ROWS_PER_BLOCK;

    // Each stage covers 2 input rows (1024 floats); thread tid owns floats
    // [tid*4, tid*4+4) of that stage.
    const int y_local = tid >> 7;             // 0 or 1: row within the stage pair
    const int x4      = (tid & 127) << 2;     // input column of first element

    const float* gin = x + ((size_t)bc * HF + row0) * WF + tid * 4;
    float* gout_base = out + (size_t)bc * HT * WT;

    const unsigned lds0 = lds_byte_addr(&smem[0][tid]);
    const unsigned lds1 = lds_byte_addr(&smem[1][tid]);

    // Prefetch stage 0 into buffer 0.
    async_copy_b128(lds0, gin);

#pragma unroll
    for (int s = 0; s < STAGES; ++s) {
        if (s + 1 < STAGES) {
            // Prefetch next stage into the other buffer, then wait for the
            // current stage only (outstanding goes 2 -> 1).
            async_copy_b128((s & 1) ? lds0 : lds1, gin + (s + 1) * (2 * WF));
            WAIT_ASYNCCNT(1);
        } else {
            WAIT_ASYNCCNT(0);
        }

        // Lane-private LDS slot: no cross-thread sharing, no barrier needed.
        const float4 v = smem[s & 1][tid];

        const int grow = row0 + 2 * s + y_local;                // input row
        float* orow = gout_base + (size_t)(2 * grow) * WT + 2 * x4;

        const float4 e0 = make_float4(v.x, 0.0f, v.y, 0.0f);
        const float4 e1 = make_float4(v.z, 0.0f, v.w, 0.0f);
        const float4 zz = make_float4(0.0f, 0.0f, 0.0f, 0.0f);

        // Even output row: interleaved values; odd output row: zeros.
        // All 16B-aligned -> global_store_b128; every out element written once.
        *(float4*)(orow)          = e0;
        *(float4*)(orow + 4)      = e1;
        *(float4*)(orow + WT)     = zz;
        *(float4*)(orow + WT + 4) = zz;
    }
}

extern "C" void kernel_launch(void* const* d_in, const int* in_sizes, int n_in,
                              void* d_out, int out_size, void* d_ws, size_t ws_size,
                              hipStream_t stream) {
    (void)in_sizes; (void)n_in; (void)out_size; (void)d_ws; (void)ws_size;
    const float* x = (const float*)d_in[0];
    float* out = (float*)d_out;

    const int blocks = BC * (HF / ROWS_PER_BLOCK); // 24 * 64 = 1536
    dilution_2x_kernel<<<blocks, THREADS, 0, stream>>>(x, out);
}